// PatchDepthNet_80642305950156
// MI455X (gfx1250) — compile-verified
//
#include <hip/hip_runtime.h>
#include <hip/hip_bf16.h>
#include <math.h>

typedef __attribute__((ext_vector_type(2))) float v2f;
typedef __attribute__((ext_vector_type(8))) float v8f;

#define HWN 262144   // 512*512
#define WID 512

// ---------------- Kernel 1: F[b,h,w] = sum_c ref_feature[b,c,h,w] ----------------
__global__ __launch_bounds__(256) void pdn_reduce_c(const float* __restrict__ ref,
                                                    float* __restrict__ Fs) {
  int p = blockIdx.x * blockDim.x + threadIdx.x;   // 0 .. 2*HWN-1
  int b = p >> 18;
  int pix = p & (HWN - 1);
  const float* base = ref + (size_t)b * 32 * HWN + pix;
  float s = 0.f;
#pragma unroll
  for (int c = 0; c < 32; ++c) s += base[(size_t)c * HWN];
  Fs[p] = s;
}

// axis tap: grid_sample coords ix = t*W/(W-1) - 0.5, border clamp
__device__ __forceinline__ void axis_tap(float t, int& c0, int& c1, float& fr) {
  float i = t * (512.0f / 511.0f) - 0.5f;
  float f0 = floorf(i);
  fr = i - f0;
  int i0 = (int)f0;
  c0 = min(max(i0, 0), 511);
  c1 = min(max(i0 + 1, 0), 511);
}

// ---------------- Kernel 2: fused geometry + sampling + WMMA MLP ----------------
__global__ __launch_bounds__(256) void pdn_main(
    const float* __restrict__ nrm, const float* __restrict__ depth,
    const float* __restrict__ intri,
    const float* __restrict__ w0, const float* __restrict__ s0, const float* __restrict__ b0,
    const float* __restrict__ w1, const float* __restrict__ s1, const float* __restrict__ b1,
    const float* __restrict__ w2, const float* __restrict__ b2,
    const float* __restrict__ Fsum, float* __restrict__ out) {
  int tid = blockIdx.x * blockDim.x + threadIdx.x;
  int b = tid >> 18;
  int w = tid & 511;
  int h = (tid >> 9) & 511;
  int lane = threadIdx.x & 31;

  float fx = intri[b * 9 + 0], cx = intri[b * 9 + 2];
  float fy = intri[b * 9 + 4], cy = intri[b * 9 + 5];
  float rfx = 1.0f / fx, rfy = 1.0f / fy;

  const float* nb0 = nrm + (size_t)b * 3 * HWN;
  int cidx = (h << 9) + w;
  size_t dbase = (size_t)b * 8 * HWN + (size_t)cidx;
  __builtin_prefetch(depth + dbase, 0, 0);   // global_prefetch_b8

  float nx = nb0[cidx], ny = nb0[cidx + HWN], nz = nb0[cidx + 2 * HWN];

  // x axis states for ox in {-1,0,+1}; y axis states for oy in {0,+1}
  int xc0[3], xc1[3]; float xfr[3], xsv[3];
#pragma unroll
  for (int k = 0; k < 3; ++k) {
    axis_tap((float)(w + k - 1), xc0[k], xc1[k], xfr[k]);
    xsv[k] = (1.0f - xfr[k]) * (float)xc0[k] + xfr[k] * (float)xc1[k];
  }
  int yc0[2], yc1[2]; float yfr[2], ysv[2];
#pragma unroll
  for (int k = 0; k < 2; ++k) {
    axis_tap((float)(h + k), yc0[k], yc1[k], yfr[k]);
    ysv[k] = (1.0f - yfr[k]) * (float)yc0[k] + yfr[k] * (float)yc1[k];
  }

  // unique offsets: center(0,0), left(0,-1), right(0,+1), down(+1,0)x2
  const int   xi_of[4] = {1, 0, 2, 1};
  const int   yi_of[4] = {0, 0, 0, 1};
  const float ml_of[4] = {1.f, 1.f, 1.f, 2.f};
  const float* Fb = Fsum + (size_t)b * HWN;

  float denv[4], simv[4], sfv[4];
#pragma unroll
  for (int i = 0; i < 4; ++i) {
    int xi = xi_of[i], yi = yi_of[i];
    float u = (xsv[xi] - cx) * rfx;
    float v = (ysv[yi] - cy) * rfy;
    denv[i] = nx * u + ny * v + nz;
    float wx1 = xfr[xi], wx0 = 1.0f - wx1;
    float wy1 = yfr[yi], wy0 = 1.0f - wy1;
    int i00 = (yc0[yi] << 9) + xc0[xi];
    int i01 = (yc0[yi] << 9) + xc1[xi];
    int i10 = (yc1[yi] << 9) + xc0[xi];
    int i11 = (yc1[yi] << 9) + xc1[xi];
    float w00 = wy0 * wx0, w01 = wy0 * wx1, w10 = wy1 * wx0, w11 = wy1 * wx1;
    float sxn = w00 * nb0[i00] + w01 * nb0[i01] + w10 * nb0[i10] + w11 * nb0[i11];
    float syn = w00 * nb0[i00 + HWN] + w01 * nb0[i01 + HWN] +
                w10 * nb0[i10 + HWN] + w11 * nb0[i11 + HWN];
    float szn = w00 * nb0[i00 + 2 * HWN] + w01 * nb0[i01 + 2 * HWN] +
                w10 * nb0[i10 + 2 * HWN] + w11 * nb0[i11 + 2 * HWN];
    sfv[i] = w00 * Fb[i00] + w01 * Fb[i01] + w10 * Fb[i10] + w11 * Fb[i11];
    float dt  = sxn * nx + syn * ny + szn * nz;
    float na  = sqrtf(sxn * sxn + syn * syn + szn * szn);
    float nbv = sqrtf(nx * nx + ny * ny + nz * nz);
    simv[i] = dt / (fmaxf(na, 1e-8f) * fmaxf(nbv, 1e-8f));
  }

  float num = denv[0];
  float ssum = 0.f, Sacc = 0.f;
#pragma unroll
  for (int i = 0; i < 4; ++i) {
    float dd = denv[i];
    if (fabsf(dd) < 1e-8f) dd = (dd >= 0.f) ? 1e-8f : -1e-8f;
    float wn = num / dd;
    if (isnan(wn) || isinf(wn)) wn = 1.0f;
    ssum += ml_of[i] * simv[i];
    Sacc += ml_of[i] * simv[i] * wn * sfv[i];
  }
  float Sc = (Sacc / ssum) * (1.0f / 32.0f);   // sim_vol = depth * Sc

  float dep[8];
#pragma unroll
  for (int d = 0; d < 8; ++d) dep[d] = depth[dbase + (size_t)d * HWN];

  // ------------- WMMA MLP: h = relu(w1f @ h0 + b1); logit = w2.h + b2 -------------
  // Wave handles 32 pixels: group A = lanes' pixels 0..15, group B = 16..31.
  int j = lane & 15;
  int dlt = (lane < 16) ? 0 : 2;   // K-offset of this lane half in A/B layouts
  float ScA = __shfl(Sc, j, 32);
  float ScB = __shfl(Sc, j + 16, 32);
  float depA[8], depB[8];
#pragma unroll
  for (int d = 0; d < 8; ++d) {
    depA[d] = __shfl(dep[d], j, 32);
    depB[d] = __shfl(dep[d], j + 16, 32);
  }

  // per-lane layer-0 constants for the K slots this lane supplies in B
  float c0v[8], c1v[8];
#pragma unroll
  for (int q = 0; q < 4; ++q) {
    int kb = 4 * q + dlt;
    c0v[2 * q]     = w0[kb] * s0[kb];         c1v[2 * q]     = b0[kb];
    c0v[2 * q + 1] = w0[kb + 1] * s0[kb + 1]; c1v[2 * q + 1] = b0[kb + 1];
  }
  // A matrix (w1 * s1, rows 8..15 zero-padded): lane holds row M=lane&15
  int M = j;
  float a0[4], a1[4];
#pragma unroll
  for (int q = 0; q < 4; ++q) {
    int kb = 4 * q + dlt;
    a0[q] = (M < 8) ? w1[M * 16 + kb] * s1[M] : 0.0f;
    a1[q] = (M < 8) ? w1[M * 16 + kb + 1] * s1[M] : 0.0f;
  }
  float b1v[8], w2v[8];
#pragma unroll
  for (int o = 0; o < 8; ++o) { b1v[o] = b1[o]; w2v[o] = w2[o]; }
  float b2v = b2[0];

  float vmA = 0.f, vmB = 0.f;   // running max over d of sigmoid (sigmoid > 0)
#pragma unroll
  for (int d = 0; d < 8; ++d) {
    float xA = ScA * depA[d];
    float xB = ScB * depB[d];
    v8f accA = {0.f, 0.f, 0.f, 0.f, 0.f, 0.f, 0.f, 0.f};
    v8f accB = {0.f, 0.f, 0.f, 0.f, 0.f, 0.f, 0.f, 0.f};
#pragma unroll
    for (int q = 0; q < 4; ++q) {
      v2f av = {a0[q], a1[q]};
      v2f bA = {fmaxf(c0v[2 * q] * xA + c1v[2 * q], 0.f),
                fmaxf(c0v[2 * q + 1] * xA + c1v[2 * q + 1], 0.f)};
      accA = __builtin_amdgcn_wmma_f32_16x16x4_f32(false, av, false, bA,
                                                   (short)0, accA, false, false);
      v2f bB = {fmaxf(c0v[2 * q] * xB + c1v[2 * q], 0.f),
                fmaxf(c0v[2 * q + 1] * xB + c1v[2 * q + 1], 0.f)};
      accB = __builtin_amdgcn_wmma_f32_16x16x4_f32(false, av, false, bB,
                                                   (short)0, accB, false, false);
    }
    // rows 0..7 of D live in VGPRs 0..7 of lanes 0..15 (element = lane index)
    float sA = b2v, sB = b2v;
#pragma unroll
    for (int o = 0; o < 8; ++o) {
      sA += w2v[o] * fmaxf(accA[o] + b1v[o], 0.f);
      sB += w2v[o] * fmaxf(accB[o] + b1v[o], 0.f);
    }
    float vwA = 1.0f / (1.0f + __expf(-sA));
    float vwB = 1.0f / (1.0f + __expf(-sB));
    vmA = fmaxf(vmA, vwA);
    vmB = fmaxf(vmB, vwB);
  }

  // lane<16 owns pixel 'lane' (group A); lane>=16 takes group B result from lane-16
  float fromB = __shfl(vmB, lane & 15, 32);
  float vwm = (lane < 16) ? vmA : fromB;

#pragma unroll
  for (int d = 0; d < 8; ++d)
    out[dbase + (size_t)d * HWN] = dep[d] * Sc * vwm;
}

extern "C" void kernel_launch(void* const* d_in, const int* in_sizes, int n_in,
                              void* d_out, int out_size, void* d_ws, size_t ws_size,
                              hipStream_t stream) {
  const float* ref   = (const float*)d_in[0];
  const float* nrm   = (const float*)d_in[1];
  const float* depth = (const float*)d_in[2];
  const float* intri = (const float*)d_in[3];
  const float* w0 = (const float*)d_in[4];
  const float* s0 = (const float*)d_in[5];
  const float* b0 = (const float*)d_in[6];
  const float* w1 = (const float*)d_in[7];
  const float* s1 = (const float*)d_in[8];
  const float* b1 = (const float*)d_in[9];
  const float* w2 = (const float*)d_in[10];
  const float* b2 = (const float*)d_in[11];
  float* out = (float*)d_out;
  float* Fs  = (float*)d_ws;   // B*H*W floats = 2 MB of scratch

  int pixels = 2 * HWN;        // B*H*W = 524288
  pdn_reduce_c<<<pixels / 256, 256, 0, stream>>>(ref, Fs);
  pdn_main<<<pixels / 256, 256, 0, stream>>>(nrm, depth, intri,
                                             w0, s0, b0, w1, s1, b1, w2, b2,
                                             Fs, out);
}